// MultiTaskLoss_58866821759444
// MI455X (gfx1250) — compile-verified
//
#include <hip/hip_runtime.h>
#include <math.h>

// MI455X / gfx1250, wave32. One block per batch element; 12 waves, each wave
// computes one 16-theta tile of the beam pattern via V_WMMA_F32_16X16X4_F32
// using the separable antenna-grid factorization:
//   phase = x*u + y*v,  u = pi*cos(el_t)*cos(az_b), v = pi*cos(el_t)*sin(az_b)
//   G[t,y] = sum_x w[y,x] e^{i x u_t}   -> two f32 WMMA GEMMs (cos part, sin part)
//   S[t]   = sum_y e^{i y v_t} G[t,y]   -> VALU + half-wave shuffle reduction
// All transcendentals use the hardware units (v_sin_f32/v_cos_f32/v_log_f32):
// args are bounded (<= 15*pi), so native range reduction is valid, and this is
// the throughput-limiting op class per the roofline (memory is only ~1MB).

typedef __attribute__((ext_vector_type(2))) float v2f;
typedef __attribute__((ext_vector_type(8))) float v8f;

#define NB      1024
#define NA      256
#define NT      181
#define NTILES  12
#define THREADS1 (NTILES * 32)

__device__ __forceinline__ float signf3(float x) {
  return (x > 0.f) ? 1.f : ((x < 0.f) ? -1.f : 0.f);
}

__global__ __launch_bounds__(THREADS1)
void beam_metrics_kernel(const float* __restrict__ weights,   // [B, 256]
                         const float* __restrict__ tgt_phi,   // [B]
                         float* __restrict__ contrib)         // [B] per-batch metric loss
{
  __shared__ float lw[NA];    // weights for this batch, w[y*16+x]
  __shared__ float lp[192];   // |S|^2 then dB, per theta (padded)

  const int b    = blockIdx.x;
  const int tid  = threadIdx.x;
  const int lane = tid & 31;
  const int wave = tid >> 5;

  if (tid < NA) lw[tid] = weights[b * NA + tid];
  __syncthreads();

  const float PI  = 3.14159265358979323846f;
  const float DEG = 0.5f * PI / 180.f;        // theta[t] = 0.5*t degrees

  const float az = tgt_phi[b] * (PI / 180.f);
  float saz, caz;
  __sincosf(az, &saz, &caz);

  const int tbase = wave * 16;
  const int m  = lane & 15;                   // A row (theta within tile) and B column (y)
  const int hi = (lane >> 4) << 1;            // K sub-offset: 0 for lanes 0-15, 2 for 16-31

  // u for this lane's A-matrix row: u = pi * cos(el) * cos(az)
  const float elA = (float)(tbase + m) * DEG;
  const float u   = PI * __cosf(elA) * caz;

  // Build A fragments (cos / sin of x*u) and B fragments (weights) for K = 0..15 in steps of 4.
  v2f aC[4], aS[4], bF[4];
  #pragma unroll
  for (int j = 0; j < 4; ++j) {
    const int   k  = 4 * j + hi;
    const float x0 = (float)k;
    float s0, c0, s1, c1;
    __sincosf(x0 * u,        &s0, &c0);
    __sincosf((x0 + 1.f) * u, &s1, &c1);
    aC[j].x = c0; aC[j].y = c1;
    aS[j].x = s0; aS[j].y = s1;
    bF[j].x = lw[m * 16 + k];       // B[x=k,   y=m] = w[y*16+x]
    bF[j].y = lw[m * 16 + k + 1];   // B[x=k+1, y=m]
  }

  // Two 16x16x16 f32 GEMMs (shared B): Cg[t,y] = sum_x w[y,x] cos(x u_t), Sg likewise with sin.
  v8f Cg = {0.f,0.f,0.f,0.f,0.f,0.f,0.f,0.f};
  v8f Sg = {0.f,0.f,0.f,0.f,0.f,0.f,0.f,0.f};
  #pragma unroll
  for (int j = 0; j < 4; ++j)
    Cg = __builtin_amdgcn_wmma_f32_16x16x4_f32(false, aC[j], false, bF[j],
                                               (short)0, Cg, false, false);
  #pragma unroll
  for (int j = 0; j < 4; ++j)
    Sg = __builtin_amdgcn_wmma_f32_16x16x4_f32(false, aS[j], false, bF[j],
                                               (short)0, Sg, false, false);

  // Stage 2: S[t] = sum_y e^{i y v_t} (Cg + i Sg); reduce over y (16 lanes per half-wave).
  #pragma unroll
  for (int r = 0; r < 8; ++r) {
    const int   trow = tbase + r + ((lane >= 16) ? 8 : 0);   // C/D layout: VGPR r -> rows r, r+8
    const float elr  = (float)trow * DEG;
    const float vr   = PI * __cosf(elr) * saz;
    float sy, cy;
    __sincosf((float)m * vr, &sy, &cy);
    const float gre = Cg[r], gim = Sg[r];
    float pre = gre * cy - gim * sy;
    float pim = gim * cy + gre * sy;
    #pragma unroll
    for (int msk = 1; msk <= 8; msk <<= 1) {
      pre += __shfl_xor(pre, msk, 32);
      pim += __shfl_xor(pim, msk, 32);
    }
    if (m == 0 && trow < NT) lp[trow] = pre * pre + pim * pim;
  }
  __syncthreads();

  // Metrics: small serial scans over 181 entries (negligible), faithful to jnp semantics.
  if (tid == 0) {
    float pmax = -1e30f; int mainIdx = 0;
    for (int t = 0; t < NT; ++t) {
      const float p = lp[t];
      if (p > pmax) { pmax = p; mainIdx = t; }   // first occurrence, like jnp.argmax
    }
    const float inv = 1.f / pmax;

    int f3 = -1, l3 = -1;
    for (int t = 0; t < NT; ++t) {
      const float db = 10.f * __log10f(lp[t] * inv + 1e-10f);
      lp[t] = db;
      if (db >= -3.f) { if (f3 < 0) f3 = t; l3 = t; }
    }
    const float bwv = (f3 >= 0) ? (float)(l3 - f3) : 1.f;

    int fn = -1;
    float sprev = signf3(lp[0] + 20.f);
    for (int j = 0; j < NT - 1; ++j) {
      const float snext = signf3(lp[j + 1] + 20.f);
      if ((snext != sprev) && (j >= mainIdx) && (fn < 0)) fn = j;
      sprev = snext;
    }
    const bool hasNull = (fn >= 0);
    const int  fni = hasNull ? fn : 0;
    float tmax = -1e30f;
    for (int t = fni; t < NT; ++t) tmax = fmaxf(tmax, lp[t]);
    const float sll  = hasNull ? ((fni < NT - 1) ? tmax : -30.f) : -20.f;
    const float asll = fabsf(sll);

    const float c1 = bwv - 10.f, c2 = asll - 30.f;
    contrib[b] = c1 * c1 * 0.01f + c2 * c2 * 0.01f;
  }
}

__global__ __launch_bounds__(256)
void loss_reduce_kernel(const float* __restrict__ pred_theta,
                        const float* __restrict__ pred_phi,
                        const float* __restrict__ tgt_theta,
                        const float* __restrict__ tgt_phi,
                        const float* __restrict__ beam_props,   // [B,3]
                        const float* __restrict__ part_w,       // [B,8]
                        const float* __restrict__ contrib,      // [B]
                        float* __restrict__ out)                // [5]
{
  __shared__ float s0[256], s1[256], s2[256], s3[256], s4[256];
  const int tid = threadIdx.x;
  float a0 = 0.f, a1 = 0.f, a2 = 0.f, a3 = 0.f, a4 = 0.f;

  for (int b = tid; b < NB; b += 256) {
    const float dt = pred_theta[b] - tgt_theta[b];
    const float dp = pred_phi[b]   - tgt_phi[b];
    a0 += dt * dt;
    a1 += dp * dp;
    a2 += contrib[b];
    a3 += beam_props[b * 3 + 2];
  }
  for (int i = tid; i < NB * 8; i += 256) {
    const float pw = part_w[i];
    a4 += pw * __logf(pw + 1e-8f);
  }
  s0[tid] = a0; s1[tid] = a1; s2[tid] = a2; s3[tid] = a3; s4[tid] = a4;
  __syncthreads();
  for (int off = 128; off > 0; off >>= 1) {
    if (tid < off) {
      s0[tid] += s0[tid + off];
      s1[tid] += s1[tid + off];
      s2[tid] += s2[tid + off];
      s3[tid] += s3[tid + off];
      s4[tid] += s4[tid + off];
    }
    __syncthreads();
  }
  if (tid == 0) {
    const float angle = s0[0] / (float)NB + s1[0] / (float)NB;
    const float bm    = s2[0] / (float)NB;
    const float snr   = -(s3[0] / (float)NB);
    const float part  = -(s4[0] / (float)(NB * 8));
    out[0] = angle + 0.5f * bm + 0.3f * snr + 0.2f * part;
    out[1] = angle;
    out[2] = bm;
    out[3] = snr;
    out[4] = part;
  }
}

extern "C" void kernel_launch(void* const* d_in, const int* in_sizes, int n_in,
                              void* d_out, int out_size, void* d_ws, size_t ws_size,
                              hipStream_t stream) {
  (void)in_sizes; (void)n_in; (void)out_size; (void)ws_size;
  const float* pred_theta = (const float*)d_in[0];
  const float* pred_phi   = (const float*)d_in[1];
  const float* weights    = (const float*)d_in[2];
  const float* beam_props = (const float*)d_in[3];
  const float* part_w     = (const float*)d_in[4];
  const float* tgt_theta  = (const float*)d_in[5];
  const float* tgt_phi    = (const float*)d_in[6];
  float* out     = (float*)d_out;
  float* contrib = (float*)d_ws;   // 1024 floats of scratch

  beam_metrics_kernel<<<NB, THREADS1, 0, stream>>>(weights, tgt_phi, contrib);
  loss_reduce_kernel<<<1, 256, 0, stream>>>(pred_theta, pred_phi, tgt_theta, tgt_phi,
                                            beam_props, part_w, contrib, out);
}